// TN_Model_27608049779367
// MI455X (gfx1250) — compile-verified
//
#include <hip/hip_runtime.h>

// MPS classifier: chain of 784 32x32 fp32 matmuls per batch element (128),
// done with V_WMMA_F32_16X16X4_F32 (the only fp32 WMMA path on CDNA5).
// One wave (32 threads, wave32) per block; running product lives in LDS.

typedef __attribute__((ext_vector_type(2))) float v2f;
typedef __attribute__((ext_vector_type(8))) float v8f;

#define BOND 32
#define NUM_SITES 784
#define NSEG 8
#define SEG_LEN (NUM_SITES / NSEG)   // 98
#define NBATCH 128
#define NCLASS 10

// One chain step: sacc (32x32, row-major in LDS) <- sacc @ B,
// where B = b0 + xv * b1 (AFFINE) or B = b0 (!AFFINE), b0/b1 row-major 32x32.
template <bool AFFINE>
__device__ __forceinline__ void mm_step(float* sacc,
                                        const float* __restrict__ b0p,
                                        const float* __restrict__ b1p,
                                        float xv) {
    const int lane = threadIdx.x;        // 0..31
    const int lo = lane & 15;
    const int hi = lane >> 4;

    v8f c00 = {}, c01 = {}, c10 = {}, c11 = {};

#pragma unroll
    for (int kb = 0; kb < 8; ++kb) {
        const int k0 = 4 * kb + 2 * hi;  // even -> 8B aligned in LDS

        // A slices (WMMA A layout): lane holds A(m=lo(+16), k=k0..k0+1)
        v2f a0 = *(const v2f*)(sacc + lo * BOND + k0);
        v2f a1 = *(const v2f*)(sacc + (16 + lo) * BOND + k0);

        // B slices (WMMA B layout): lane holds B(k=k0..k0+1, n=lo(+16))
        const int off0 = (k0 + 0) * BOND + lo;
        const int off1 = (k0 + 1) * BOND + lo;
        float e00 = b0p[off0];        // col-block 0, r=0
        float e01 = b0p[off1];        // col-block 0, r=1
        float e10 = b0p[off0 + 16];   // col-block 1, r=0
        float e11 = b0p[off1 + 16];   // col-block 1, r=1
        if (AFFINE) {
            e00 = fmaf(xv, b1p[off0],      e00);
            e01 = fmaf(xv, b1p[off1],      e01);
            e10 = fmaf(xv, b1p[off0 + 16], e10);
            e11 = fmaf(xv, b1p[off1 + 16], e11);
        }
        v2f b0; b0.x = e00; b0.y = e01;
        v2f b1; b1.x = e10; b1.y = e11;

        c00 = __builtin_amdgcn_wmma_f32_16x16x4_f32(false, a0, false, b0, (short)0, c00, false, false);
        c01 = __builtin_amdgcn_wmma_f32_16x16x4_f32(false, a0, false, b1, (short)0, c01, false, false);
        c10 = __builtin_amdgcn_wmma_f32_16x16x4_f32(false, a1, false, b0, (short)0, c10, false, false);
        c11 = __builtin_amdgcn_wmma_f32_16x16x4_f32(false, a1, false, b1, (short)0, c11, false, false);
    }

    // Scatter C tiles back to LDS row-major. Same-wave LDS ops are in-order,
    // so the next iteration's A loads observe these stores without a barrier.
#pragma unroll
    for (int r = 0; r < 8; ++r) {
        const int m = r + 8 * hi;
        sacc[m * BOND + lo]              = c00[r];
        sacc[m * BOND + 16 + lo]         = c01[r];
        sacc[(16 + m) * BOND + lo]       = c10[r];
        sacc[(16 + m) * BOND + 16 + lo]  = c11[r];
    }
}

// Kernel 1: each wave reduces one segment of 98 site matrices for one batch
// element. grid = 128 * 8 blocks of 32 threads.
__global__ void __launch_bounds__(32)
tn_segment_kernel(const float* __restrict__ x,   // (128, 784)
                  const float* __restrict__ w,   // (2, 784, 32, 32)
                  float* __restrict__ ws) {      // (128, 8, 32, 32)
    __shared__ float sacc[BOND * BOND];
    const int blk  = blockIdx.x;
    const int n    = blk >> 3;
    const int s    = blk & 7;
    const int lane = threadIdx.x;
    const int site0 = s * SEG_LEN;

    // Init acc = M[site0] = W0[site0] + x[n, p(site0)] * W1[site0]
    {
        const int p = (site0 + 392) % NUM_SITES;
        const float xv = x[n * NUM_SITES + p];
        const float* w0 = w + (size_t)site0 * (BOND * BOND);
        const float* w1 = w + (size_t)(NUM_SITES + site0) * (BOND * BOND);
#pragma unroll
        for (int t = 0; t < 32; ++t) {
            const int idx = t * 32 + lane;   // coalesced
            sacc[idx] = fmaf(xv, w1[idx], w0[idx]);
        }
    }

    // Left-to-right chain over the segment.
#pragma unroll 1
    for (int i = site0 + 1; i < site0 + SEG_LEN; ++i) {
        const int p = (i + 392) % NUM_SITES;
        const float xv = x[n * NUM_SITES + p];
        mm_step<true>(sacc,
                      w + (size_t)i * (BOND * BOND),
                      w + (size_t)(NUM_SITES + i) * (BOND * BOND),
                      xv);
    }

    float* dst = ws + (size_t)blk * (BOND * BOND);
#pragma unroll
    for (int t = 0; t < 32; ++t) {
        const int idx = t * 32 + lane;
        dst[idx] = sacc[idx];
    }
}

// Kernel 2: reduce the 8 segment matrices per batch, then contract with
// `middle` and emit out[n, 0..9]. grid = 128 blocks of 32 threads.
__global__ void __launch_bounds__(32)
tn_final_kernel(const float* __restrict__ ws,      // (128, 8, 32, 32)
                const float* __restrict__ middle,  // (10, 32, 32)
                float* __restrict__ out) {         // (128, 10)
    __shared__ float sacc[BOND * BOND];
    const int n    = blockIdx.x;
    const int lane = threadIdx.x;
    const float* seg = ws + (size_t)n * NSEG * (BOND * BOND);

#pragma unroll
    for (int t = 0; t < 32; ++t) {
        const int idx = t * 32 + lane;
        sacc[idx] = seg[idx];
    }

#pragma unroll 1
    for (int s = 1; s < NSEG; ++s) {
        mm_step<false>(sacc, seg + (size_t)s * (BOND * BOND), nullptr, 0.0f);
    }

    // out[n,o] = sum_{k,l} acc(k,l) * middle[o,l,k]; lane handles k = lane.
    float row[BOND];
#pragma unroll
    for (int l = 0; l < BOND; ++l) row[l] = sacc[lane * BOND + l];

#pragma unroll 1
    for (int o = 0; o < NCLASS; ++o) {
        float p = 0.0f;
#pragma unroll
        for (int l = 0; l < BOND; ++l)
            p = fmaf(row[l], middle[o * (BOND * BOND) + l * BOND + lane], p);
#pragma unroll
        for (int off = 16; off >= 1; off >>= 1)
            p += __shfl_xor(p, off, 32);
        if (lane == 0) out[n * NCLASS + o] = p;
    }
}

extern "C" void kernel_launch(void* const* d_in, const int* in_sizes, int n_in,
                              void* d_out, int out_size, void* d_ws, size_t ws_size,
                              hipStream_t stream) {
    (void)in_sizes; (void)n_in; (void)out_size; (void)ws_size;
    const float* x      = (const float*)d_in[0];  // (128,28,28) f32
    const float* w      = (const float*)d_in[1];  // (2,784,32,32) f32
    const float* middle = (const float*)d_in[2];  // (10,32,32) f32
    float* out = (float*)d_out;                   // (128,10) f32
    float* ws  = (float*)d_ws;                    // needs 128*8*1024*4 = 4 MB

    tn_segment_kernel<<<dim3(NBATCH * NSEG), dim3(32), 0, stream>>>(x, w, ws);
    tn_final_kernel<<<dim3(NBATCH), dim3(32), 0, stream>>>(ws, middle, out);
}